// RoiPooling_15221364097271
// MI455X (gfx1250) — compile-verified
//
#include <hip/hip_runtime.h>
#include <hip/hip_bf16.h>

#define POOL 7
#define BLOCK 64
#define DEPTH 4   // async pipeline depth (stage buffers)

#define AS1 __attribute__((address_space(1)))
#define AS3 __attribute__((address_space(3)))

#if defined(__HIP_DEVICE_COMPILE__)
#if __has_builtin(__builtin_amdgcn_global_load_async_to_lds_b32) && \
    __has_builtin(__builtin_amdgcn_s_wait_asynccnt)
#define HAVE_ASYNC_LDS 1
#endif
#endif

// One block per (roi, channel). Threads cooperatively max-reduce the ROI
// region into 49 bins held in LDS (uint atomic max on an order-preserving
// key), then 49 coalesced stores. Pixel traffic is staged global->LDS with
// async-to-LDS loads (ASYNCcnt), pipelined 4 chunks (DEPTH) deep per wave.
__global__ __launch_bounds__(BLOCK) void roi_pool_kernel(
    const float* __restrict__ fmap, const int* __restrict__ roi,
    const int* __restrict__ inner, float* __restrict__ out,
    int B, int C, int H, int W)
{
    const int t   = threadIdx.x;
    const int blk = blockIdx.x;
    const int r   = blk / C;       // roi index
    const int c   = blk - r * C;   // channel

    // Replicate jnp.searchsorted(cumsum(inner), r-1, 'right') with clip.
    int img = 0;
    {
        int cs = 0;
        for (int j = 0; j < B; ++j) { cs += inner[j]; if (cs <= r - 1) ++img; }
        if (img > B - 1) img = B - 1;
    }

    const int xmin = roi[r * 4 + 0], ymin = roi[r * 4 + 1];
    const int xmax = roi[r * 4 + 2], ymax = roi[r * 4 + 3];
    const int rh   = max(ymax - ymin, 1);
    const int rw   = max(xmax - xmin, 1);
    const int regH = max(ymax - ymin, 0);
    const int regW = max(xmax - xmin, 0);
    const int npix = regH * regW;

    __shared__ unsigned bins[POOL * POOL];
    __shared__ int bi_tab[56];
    __shared__ int bj_tab[56];

    if (t < POOL * POOL) bins[t] = 0x007FFFFFu;              // key(-inf)
    if (t < regH) bi_tab[t] = ((t + 1) * POOL + rh - 1) / rh - 1;
    if (t < regW) bj_tab[t] = ((t + 1) * POOL + rw - 1) / rw - 1;
    __syncthreads();

    const float* __restrict__ base =
        fmap + ((size_t)img * C + c) * (size_t)(H * W);

#if defined(HAVE_ASYNC_LDS)
    // Async global->LDS pipeline, depth DEPTH. Each lane owns one slot per
    // buffer, so only s_wait_asynccnt (per-wave, in-order) is needed before
    // consuming -- no block barrier. Tail lanes clamp to a valid pixel so
    // every chunk issues with full EXEC (uniform ASYNCcnt accounting).
    __shared__ float stage[DEPTH][BLOCK];
    const int nchunk = (npix + BLOCK - 1) / BLOCK;   // npix >= 0 here
    auto issue = [&](int k) {
        int i = k * BLOCK + t;
        if (i >= npix) i = npix - 1;
        if (i < 0) i = 0;
        const int dy = (int)((unsigned)i / (unsigned)regW);
        const int dx = i - dy * regW;
        const float* gp = base + (ymin + dy) * W + (xmin + dx);
        __builtin_amdgcn_global_load_async_to_lds_b32(
            (AS1 int*)gp, (AS3 int*)&stage[k % DEPTH][t], 0, 0);
    };
    issue(0);
    if (1 < nchunk) issue(1);
    if (2 < nchunk) issue(2);
    for (int k = 0; k < nchunk; ++k) {
        const int rem = nchunk - 1 - k;   // chunks still in flight after k
        if (k + DEPTH - 1 < nchunk) {
            issue(k + DEPTH - 1);
            __builtin_amdgcn_s_wait_asynccnt(DEPTH - 1);
        } else if (rem >= 2) {
            __builtin_amdgcn_s_wait_asynccnt(2);
        } else if (rem == 1) {
            __builtin_amdgcn_s_wait_asynccnt(1);
        } else {
            __builtin_amdgcn_s_wait_asynccnt(0);
        }
        const int i = k * BLOCK + t;
        if (i < npix) {
            const float v  = stage[k % DEPTH][t];
            const int dy   = (int)((unsigned)i / (unsigned)regW);
            const int dx   = i - dy * regW;
            const int b    = bi_tab[dy] * POOL + bj_tab[dx];
            unsigned bits  = __float_as_uint(v);
            unsigned key   = (bits & 0x80000000u) ? ~bits : (bits | 0x80000000u);
            atomicMax(&bins[b], key);
        }
    }
#else
    for (int i = t; i < npix; i += BLOCK) {
        const int dy  = (int)((unsigned)i / (unsigned)regW);
        const int dx  = i - dy * regW;
        const float v = base[(ymin + dy) * W + (xmin + dx)];
        const int b   = bi_tab[dy] * POOL + bj_tab[dx];
        unsigned bits = __float_as_uint(v);
        unsigned key  = (bits & 0x80000000u) ? ~bits : (bits | 0x80000000u);
        atomicMax(&bins[b], key);
    }
#endif

    __syncthreads();
    if (t < POOL * POOL) {
        const unsigned key = bins[t];
        float v;
        if (key == 0x007FFFFFu) {
            v = 0.0f;  // empty bin: reference maps -inf -> 0
        } else {
            unsigned bits = (key & 0x80000000u) ? (key ^ 0x80000000u) : ~key;
            v = __uint_as_float(bits);
        }
        out[((size_t)r * C + c) * (size_t)(POOL * POOL) + t] = v;
    }
}

extern "C" void kernel_launch(void* const* d_in, const int* in_sizes, int n_in,
                              void* d_out, int out_size, void* d_ws, size_t ws_size,
                              hipStream_t stream) {
    const float* fmap  = (const float*)d_in[0];
    const int*   roi   = (const int*)d_in[1];
    const int*   inner = (const int*)d_in[2];
    float*       out   = (float*)d_out;

    const int B    = in_sizes[2];            // 8
    const int nroi = in_sizes[1] / 4;        // 256
    const int H = 56, W = 56;
    const int C = in_sizes[0] / (B * H * W); // 256

    dim3 grid((unsigned)(nroi * C));
    roi_pool_kernel<<<grid, BLOCK, 0, stream>>>(fmap, roi, inner, out, B, C, H, W);
}